// SAGENet_35038343201309
// MI455X (gfx1250) — compile-verified
//
#include <hip/hip_runtime.h>
#include <cstdint>

#define N_NODES 100000
#define N_EDGES 1600000
#define F_IN 256
#define DIM 32
#define NCLS 40

typedef __attribute__((ext_vector_type(2))) float v2f;
typedef __attribute__((ext_vector_type(8))) float v8f;
typedef __attribute__((ext_vector_type(4))) int   v4i;

// ---------- CDNA5 async global->LDS staging (direct-to-LDS, ASYNCcnt tracked) ----------
#if defined(__has_builtin)
#if __has_builtin(__builtin_amdgcn_global_load_async_to_lds_b128)
#define HAVE_ASYNC_LDS 1
#endif
#endif

__device__ __forceinline__ void stage16B(const float* g, float* l) {
#ifdef HAVE_ASYNC_LDS
    // signature (from hipcc diagnostic): (int4* gsrc, int4* lds, imm offset, imm cpol)
    __builtin_amdgcn_global_load_async_to_lds_b128(
        (v4i*)const_cast<float*>(g), (v4i*)l, 0, 0);
#else
    *(float4*)l = *(const float4*)g;
#endif
}

__device__ __forceinline__ void stage_wait() {
#ifdef HAVE_ASYNC_LDS
#if __has_builtin(__builtin_amdgcn_s_wait_asynccnt)
    __builtin_amdgcn_s_wait_asynccnt(0);
#else
    asm volatile("s_wait_asynccnt 0x0" ::: "memory");
#endif
#endif
}

// ---------------- degree ----------------
__global__ void k_deg(const long long* __restrict__ dst, float* __restrict__ deg) {
    int e = blockIdx.x * blockDim.x + threadIdx.x;
    if (e < N_EDGES) atomicAdd(&deg[(size_t)dst[e]], 1.0f);
}

__global__ void k_invdeg(float* __restrict__ deg) {
    int n = blockIdx.x * blockDim.x + threadIdx.x;
    if (n < N_NODES) deg[n] = 1.0f / fmaxf(deg[n], 1.0f);
}

// ---------------- GEMM1: X[N,256] @ [W1l|W1r][256,64] -> YZ1[N,64] ----------------
// block = 128 threads (4 waves), 32 rows per block. Wave w owns n-tile w (16 cols)
// and computes TWO 16x16 M-tiles sharing the same B fragments.
__global__ void k_gemm1(const float* __restrict__ x,
                        const float* __restrict__ W1l,
                        const float* __restrict__ W1r,
                        float* __restrict__ yz1) {
    __shared__ float lds[32 * 260];  // pad stride 260: banks (4r+c)%64 distinct, both tiles
    const int m0  = blockIdx.x * 32;
    const int tid = threadIdx.x;

    // stage 32x256 X tile directly into LDS (async b128 if available)
    for (int q = tid; q < 32 * 64; q += 128) {
        int row = q >> 6, c4 = q & 63;
        stage16B(x + (size_t)(m0 + row) * F_IN + c4 * 4, &lds[row * 260 + c4 * 4]);
    }
    stage_wait();
    __syncthreads();

    const int wave = tid >> 5;          // 0..3 -> n-tile
    const int lane = tid & 31;
    const int hlf  = lane >> 4;         // half-wave select
    const int r    = lane & 15;

    const float* __restrict__ W = (wave < 2) ? W1l : W1r;
    const int n = (wave & 1) * 16 + r;  // column within the 32-wide weight

    v8f acc0 = {};
    v8f acc1 = {};
    for (int k = 0; k < F_IN; k += 4) {
        // A: 16x4 f32, a[v] = A[M=r][K = 2*hlf + v]
        v2f a0 = *(const v2f*)&lds[r * 260 + k + 2 * hlf];
        v2f a1 = *(const v2f*)&lds[(16 + r) * 260 + k + 2 * hlf];
        // B: 4x16 f32, b[v] = B[K = 2*hlf + v][N = r]  (shared by both M-tiles)
        v2f b;
        b.x = W[(size_t)(k + 2 * hlf + 0) * DIM + n];
        b.y = W[(size_t)(k + 2 * hlf + 1) * DIM + n];
        acc0 = __builtin_amdgcn_wmma_f32_16x16x4_f32(false, a0, false, b,
                                                     (short)0, acc0, false, false);
        acc1 = __builtin_amdgcn_wmma_f32_16x16x4_f32(false, a1, false, b,
                                                     (short)0, acc1, false, false);
    }

    // D: d[v] = D[M = v + 8*hlf][N = r]
    const int col = wave * 16 + r;
#pragma unroll
    for (int v = 0; v < 8; ++v) {
        yz1[(size_t)(m0 + v + 8 * hlf) * 64 + col]      = acc0[v];
        yz1[(size_t)(m0 + 16 + v + 8 * hlf) * 64 + col] = acc1[v];
    }
}

// ---------------- scatter layer 1: agg1[dst,j] += y1[src,j], j=lane ----------------
__global__ void k_scatter1(const long long* __restrict__ src,
                           const long long* __restrict__ dst,
                           const float* __restrict__ yz1,
                           float* __restrict__ agg1) {
    size_t idx = (size_t)blockIdx.x * blockDim.x + threadIdx.x;
    size_t e = idx >> 5;           // wave-uniform edge id
    int j = (int)(idx & 31);
    if (e >= N_EDGES) return;
    size_t s = (size_t)src[e], d = (size_t)dst[e];
    atomicAdd(&agg1[d * DIM + j], yz1[s * 64 + j]);
}

// ---------------- h = relu(agg1*inv_deg + z1 + b1) ----------------
__global__ void k_makeh(const float* __restrict__ yz1,
                        const float* __restrict__ agg1,
                        const float* __restrict__ invdeg,
                        const float* __restrict__ b1,
                        float* __restrict__ h) {
    size_t i = (size_t)blockIdx.x * blockDim.x + threadIdx.x;
    if (i >= (size_t)N_NODES * DIM) return;
    size_t node = i >> 5;
    int j = (int)(i & 31);
    float v = agg1[i] * invdeg[node] + yz1[node * 64 + DIM + j] + b1[j];
    h[i] = fmaxf(v, 0.0f);
}

// ---------------- GEMM2: H[N,32] @ [W2l|W2r][32,80] -> YZ2[N,80] ----------------
// block = 160 threads (5 waves), 16 rows per block, wave w -> 16 output cols.
__global__ void k_gemm2(const float* __restrict__ h,
                        const float* __restrict__ W2l,
                        const float* __restrict__ W2r,
                        float* __restrict__ yz2) {
    __shared__ float lds[16 * 36];  // pad stride 36: banks (36r+c)%64 distinct for r=0..15
    const int m0  = blockIdx.x * 16;
    const int tid = threadIdx.x;

    // 16 rows x 32 floats = 128 float4 slots
    if (tid < 128) {
        int row = tid >> 3, c4 = tid & 7;
        stage16B(h + (size_t)(m0 + row) * DIM + c4 * 4, &lds[row * 36 + c4 * 4]);
    }
    stage_wait();
    __syncthreads();

    const int wave = tid >> 5;     // 0..4
    const int lane = tid & 31;
    const int hlf  = lane >> 4;
    const int r    = lane & 15;

    const int n = wave * 16 + r;   // 0..79
    const float* __restrict__ W = (n < NCLS) ? W2l : W2r;
    const int nc = (n < NCLS) ? n : (n - NCLS);

    v8f acc = {};
    for (int k = 0; k < DIM; k += 4) {
        v2f a = *(const v2f*)&lds[r * 36 + k + 2 * hlf];
        v2f b;
        b.x = W[(size_t)(k + 2 * hlf + 0) * NCLS + nc];
        b.y = W[(size_t)(k + 2 * hlf + 1) * NCLS + nc];
        acc = __builtin_amdgcn_wmma_f32_16x16x4_f32(false, a, false, b,
                                                    (short)0, acc, false, false);
    }

#pragma unroll
    for (int v = 0; v < 8; ++v)
        yz2[(size_t)(m0 + v + 8 * hlf) * 80 + n] = acc[v];
}

// ---------------- scatter layer 2: agg2[dst,j] += y2[src,j], j=0..39 ----------------
__global__ void k_scatter2(const long long* __restrict__ src,
                           const long long* __restrict__ dst,
                           const float* __restrict__ yz2,
                           float* __restrict__ agg2) {
    size_t idx = (size_t)blockIdx.x * blockDim.x + threadIdx.x;
    if (idx >= (size_t)N_EDGES * NCLS) return;
    size_t e = idx / NCLS;
    int j = (int)(idx - e * NCLS);
    size_t s = (size_t)src[e], d = (size_t)dst[e];
    atomicAdd(&agg2[d * NCLS + j], yz2[s * 80 + j]);
}

// ---------------- final: out = log_softmax(agg2*inv_deg + z2 + b2) ----------------
// one wave32 per node; lane covers class {lane, lane+32}
__global__ void k_final(const float* __restrict__ yz2,
                        const float* __restrict__ agg2,
                        const float* __restrict__ invdeg,
                        const float* __restrict__ b2,
                        float* __restrict__ out) {
    int node = blockIdx.x * 8 + (threadIdx.x >> 5);
    int lane = threadIdx.x & 31;
    if (node >= N_NODES) return;

    const float id = invdeg[node];
    const size_t a0 = (size_t)node * NCLS;
    const size_t z0 = (size_t)node * 80 + NCLS;

    float v0 = agg2[a0 + lane] * id + yz2[z0 + lane] + b2[lane];
    float v1 = -3.402823466e+38f;
    if (lane < 8)
        v1 = agg2[a0 + 32 + lane] * id + yz2[z0 + 32 + lane] + b2[32 + lane];

    float m = fmaxf(v0, v1);
#pragma unroll
    for (int off = 16; off >= 1; off >>= 1)
        m = fmaxf(m, __shfl_xor(m, off, 32));

    float s = __expf(v0 - m) + ((lane < 8) ? __expf(v1 - m) : 0.0f);
#pragma unroll
    for (int off = 16; off >= 1; off >>= 1)
        s += __shfl_xor(s, off, 32);

    float lse = m + __logf(s);
    out[a0 + lane] = v0 - lse;
    if (lane < 8) out[a0 + 32 + lane] = v1 - lse;
}

extern "C" void kernel_launch(void* const* d_in, const int* in_sizes, int n_in,
                              void* d_out, int out_size, void* d_ws, size_t ws_size,
                              hipStream_t stream) {
    (void)in_sizes; (void)n_in; (void)out_size; (void)ws_size;
    const float*     x   = (const float*)d_in[0];
    const long long* ei  = (const long long*)d_in[1];  // int64 (2, E): src row then dst row
    const float*     W1l = (const float*)d_in[2];
    const float*     W1r = (const float*)d_in[3];
    const float*     b1  = (const float*)d_in[4];
    const float*     W2l = (const float*)d_in[5];
    const float*     W2r = (const float*)d_in[6];
    const float*     b2  = (const float*)d_in[7];
    float* out = (float*)d_out;

    const long long* src = ei;
    const long long* dst = ei + N_EDGES;

    char* ws = (char*)d_ws;
    size_t off = 0;
    auto alloc = [&](size_t nfl) { float* p = (float*)(ws + off); off += nfl * sizeof(float); return p; };
    float* yz1  = alloc((size_t)N_NODES * 64);    // [y1 | z1]
    float* agg1 = alloc((size_t)N_NODES * DIM);
    float* h    = alloc((size_t)N_NODES * DIM);
    float* yz2  = alloc((size_t)N_NODES * 80);    // [y2 | z2]
    float* agg2 = alloc((size_t)N_NODES * NCLS);
    float* deg  = alloc((size_t)N_NODES);         // becomes inv_deg in place

    // deterministic per-call zero init (graph-capture-safe memset nodes)
    (void)hipMemsetAsync(deg,  0, (size_t)N_NODES * sizeof(float), stream);
    (void)hipMemsetAsync(agg1, 0, (size_t)N_NODES * DIM  * sizeof(float), stream);
    (void)hipMemsetAsync(agg2, 0, (size_t)N_NODES * NCLS * sizeof(float), stream);

    k_deg   <<<(N_EDGES + 255) / 256, 256, 0, stream>>>(dst, deg);
    k_invdeg<<<(N_NODES + 255) / 256, 256, 0, stream>>>(deg);

    k_gemm1<<<N_NODES / 32, 128, 0, stream>>>(x, W1l, W1r, yz1);

    k_scatter1<<<(unsigned)(((size_t)N_EDGES * 32 + 255) / 256), 256, 0, stream>>>(src, dst, yz1, agg1);
    k_makeh<<<(unsigned)(((size_t)N_NODES * DIM + 255) / 256), 256, 0, stream>>>(yz1, agg1, deg, b1, h);

    k_gemm2<<<N_NODES / 16, 160, 0, stream>>>(h, W2l, W2r, yz2);

    k_scatter2<<<(unsigned)(((size_t)N_EDGES * NCLS + 255) / 256), 256, 0, stream>>>(src, dst, yz2, agg2);
    k_final<<<(N_NODES + 7) / 8, 256, 0, stream>>>(yz2, agg2, deg, b2, out);
}